// Head_56358560858532
// MI455X (gfx1250) — compile-verified
//
#include <hip/hip_runtime.h>
#include <math.h>

// Problem constants (match reference)
#define BB 128
#define CC 1024
#define NN 8192
#define MM 128
#define CHUNKS 32              // n-chunks per batch for data pass
#define ROWS_PER_CHUNK (NN / CHUNKS)   // 256
#define EPSF 1e-16f

typedef float v2f __attribute__((ext_vector_type(2)));
typedef float v4f __attribute__((ext_vector_type(4)));
typedef float v8f __attribute__((ext_vector_type(8)));

// ---------------- helpers ----------------

__device__ __forceinline__ float softplus_f(float x) {
    return (x > 20.0f) ? x : log1pf(expf(x));
}
__device__ __forceinline__ float sigmoid_f(float x) {
    return 1.0f / (1.0f + expf(-x));
}

// Deterministic block reduction: wave shfl + fixed-order LDS sum over waves.
__device__ __forceinline__ float blockReduceSum(float v, float* red, int nwaves) {
    for (int off = 16; off; off >>= 1) v += __shfl_xor(v, off, 32);
    int wid = threadIdx.x >> 5, lane = threadIdx.x & 31;
    __syncthreads();
    if (lane == 0) red[wid] = v;
    __syncthreads();
    float tot = 0.0f;
    for (int j = 0; j < nwaves; ++j) tot += red[j];
    return tot;
}
__device__ __forceinline__ float blockReduceMax(float v, float* red, int nwaves) {
    for (int off = 16; off; off >>= 1) v = fmaxf(v, __shfl_xor(v, off, 32));
    int wid = threadIdx.x >> 5, lane = threadIdx.x & 31;
    __syncthreads();
    if (lane == 0) red[wid] = v;
    __syncthreads();
    float tot = -3.4e38f;
    for (int j = 0; j < nwaves; ++j) tot = fmaxf(tot, red[j]);
    return tot;
}

// ---------------- kernel 1: k = cs @ Wk + bk  via V_WMMA_F32_16X16X4_F32 ----------------
// Grid: 64 blocks (8x8 tiles of the 128x128 output), 32 threads (one wave) each.
// ISA lane layouts (CDNA5 ISA):
//   A 16x4 f32:  lanes 0-15 hold K=0,1 in v[0],v[1];  lanes 16-31 hold K=2,3
//   B 4x16 f32:  v[0] lanes0-15 = row K=0, lanes16-31 = row K=2; v[1] = rows K=1 / K=3
//   C 16x16 f32: VGPR r: lanes0-15 -> M=r, lanes16-31 -> M=r+8
__global__ __launch_bounds__(32) void k_gemm_wmma(const float* __restrict__ cs,
                                                  const float* __restrict__ Wk,
                                                  const float* __restrict__ bk,
                                                  float* __restrict__ kout) {
    const int tile = blockIdx.x;
    const int tb = (tile >> 3) * 16;   // batch-row tile
    const int tm = (tile & 7) * 16;    // output-col tile
    const int lane = threadIdx.x;
    const int half = lane >> 4;        // 0: K 0,1  1: K 2,3
    const int l = lane & 15;

    v8f acc = {};
    const float* arow = cs + (size_t)(tb + l) * CC;
    for (int kk = 0; kk < CC; kk += 4) {
        const int ka = kk + 2 * half;
        v2f a, b;
        a[0] = arow[ka];
        a[1] = arow[ka + 1];
        b[0] = Wk[(size_t)ka * MM + tm + l];
        b[1] = Wk[(size_t)(ka + 1) * MM + tm + l];
        acc = __builtin_amdgcn_wmma_f32_16x16x4_f32(
            /*neg_a=*/false, a, /*neg_b=*/false, b,
            /*c_mod=*/(short)0, acc, /*reuse_a=*/false, /*reuse_b=*/false);
    }
    const int col = tm + l;
    const float bias = bk[col];
    #pragma unroll
    for (int r = 0; r < 8; ++r) {
        const int row = tb + r + 8 * half;
        kout[(size_t)row * MM + col] = acc[r] + bias;
    }
}

// ---------------- kernel 2: per-batch scalar heads beta, g, s(3), y ----------------
__global__ __launch_bounds__(256) void scalars_kernel(const float* __restrict__ cs,
        const float* __restrict__ Wb, const float* __restrict__ bb,
        const float* __restrict__ Wg, const float* __restrict__ bg,
        const float* __restrict__ Ws, const float* __restrict__ bs,
        const float* __restrict__ Wy, const float* __restrict__ by,
        float* __restrict__ beta, float* __restrict__ g,
        float* __restrict__ s, float* __restrict__ y) {
    __shared__ float red[32];
    const int b = blockIdx.x, t = threadIdx.x;
    float ab = 0.f, ag = 0.f, ay = 0.f, a0 = 0.f, a1 = 0.f, a2 = 0.f;
    for (int c = t; c < CC; c += 256) {
        const float x = cs[(size_t)b * CC + c];
        ab += x * Wb[c];
        ag += x * Wg[c];
        ay += x * Wy[c];
        a0 += x * Ws[c * 3 + 0];
        a1 += x * Ws[c * 3 + 1];
        a2 += x * Ws[c * 3 + 2];
    }
    ab = blockReduceSum(ab, red, 8);
    ag = blockReduceSum(ag, red, 8);
    ay = blockReduceSum(ay, red, 8);
    a0 = blockReduceSum(a0, red, 8);
    a1 = blockReduceSum(a1, red, 8);
    a2 = blockReduceSum(a2, red, 8);
    if (t == 0) {
        beta[b] = softplus_f(ab + bb[0]);
        g[b]    = sigmoid_f(ag + bg[0]);
        y[b]    = 1.0f + softplus_f(ay + by[0]);
        const float v0 = a0 + bs[0], v1 = a1 + bs[1], v2 = a2 + bs[2];
        const float m = fmaxf(v0, fmaxf(v1, v2));
        const float e0 = expf(v0 - m), e1 = expf(v1 - m), e2 = expf(v2 - m);
        const float inv = 1.0f / (e0 + e1 + e2);
        s[b * 3 + 0] = e0 * inv;
        s[b * 3 + 1] = e1 * inv;
        s[b * 3 + 2] = e2 * inv;
    }
}

// ---------------- kernel 3: knorm[b] = ||k[b]|| ----------------
__global__ __launch_bounds__(32) void knorm_kernel(const float* __restrict__ k,
                                                    float* __restrict__ knorm) {
    const int b = blockIdx.x, lane = threadIdx.x;
    const v4f kv = ((const v4f*)(k + (size_t)b * MM))[lane];
    float ss = kv.x * kv.x + kv.y * kv.y + kv.z * kv.z + kv.w * kv.w;
    for (int off = 16; off; off >>= 1) ss += __shfl_xor(ss, off, 32);
    if (lane == 0) knorm[b] = sqrtf(ss);
}

// ---------------- kernel 4: streaming pass 1 — fused dot & row-norm -> cos ----------------
// Grid: B * (N/64) blocks, 256 threads (8 waves); each wave handles 8 rows.
__global__ __launch_bounds__(256) void cos_kernel(const float* __restrict__ mem,
                                                   const float* __restrict__ k,
                                                   const float* __restrict__ knorm,
                                                   float* __restrict__ cosv) {
    const int b = blockIdx.x >> 7;                 // N/64 = 128 blocks per batch
    const int rowblk = (blockIdx.x & 127) * 64;
    const int wave = threadIdx.x >> 5;
    const int lane = threadIdx.x & 31;
    const v4f kv = ((const v4f*)(k + (size_t)b * MM))[lane];
    const float kn = knorm[b];
    const float* base = mem + (size_t)b * NN * MM;
    #pragma unroll
    for (int i = 0; i < 8; ++i) {
        const int n = rowblk + wave * 8 + i;
        const v4f* rp = (const v4f*)(base + (size_t)n * MM);
        const v4f mv = __builtin_nontemporal_load(rp + lane);   // streaming: NT hint
        float d  = mv.x * kv.x + mv.y * kv.y + mv.z * kv.z + mv.w * kv.w;
        float ss = mv.x * mv.x + mv.y * mv.y + mv.z * mv.z + mv.w * mv.w;
        for (int off = 16; off; off >>= 1) {
            d  += __shfl_xor(d, off, 32);
            ss += __shfl_xor(ss, off, 32);
        }
        if (lane == 0)
            cosv[(size_t)b * NN + n] = d / (sqrtf(ss) * kn + EPSF);
    }
}

// ---------------- kernel 5: per-batch softmax + interpolate + shift + sharpen -> w ----------------
// Grid: B blocks, 1024 threads; iw held in LDS (32 KB) for the circular shift.
__global__ __launch_bounds__(1024) void weights_kernel(const float* __restrict__ cosv,
        const float* __restrict__ pw,
        const float* __restrict__ beta, const float* __restrict__ g,
        const float* __restrict__ s, const float* __restrict__ y,
        float* __restrict__ wout) {
    __shared__ float iw[NN];
    __shared__ float red[32];
    const int b = blockIdx.x, t = threadIdx.x;
    const float be = beta[b], gg = g[b], yy = y[b];
    const float s0 = s[b * 3 + 0], s1 = s[b * 3 + 1], s2 = s[b * 3 + 2];

    float z[8];
    float mx = -3.4e38f;
    #pragma unroll
    for (int i = 0; i < 8; ++i) {
        const int n = i * 1024 + t;
        z[i] = be * cosv[(size_t)b * NN + n];
        mx = fmaxf(mx, z[i]);
    }
    mx = blockReduceMax(mx, red, 32);

    float esum = 0.f;
    #pragma unroll
    for (int i = 0; i < 8; ++i) { z[i] = expf(z[i] - mx); esum += z[i]; }
    esum = blockReduceSum(esum, red, 32);
    const float einv = 1.0f / esum;

    #pragma unroll
    for (int i = 0; i < 8; ++i) {
        const int n = i * 1024 + t;
        const float cw = z[i] * einv;
        iw[n] = gg * cw + (1.0f - gg) * pw[(size_t)b * NN + n];
    }
    __syncthreads();

    float wv[8];
    float wsum = 0.f;
    #pragma unroll
    for (int i = 0; i < 8; ++i) {
        const int n = i * 1024 + t;
        const int nm = (n == 0) ? (NN - 1) : (n - 1);
        const int np = (n == NN - 1) ? 0 : (n + 1);
        const float sw = s0 * iw[nm] + s1 * iw[n] + s2 * iw[np];
        wv[i] = powf(sw, yy);
        wsum += wv[i];
    }
    wsum = blockReduceSum(wsum, red, 32);
    const float winv = 1.0f / (wsum + EPSF);
    #pragma unroll
    for (int i = 0; i < 8; ++i)
        wout[(size_t)b * NN + i * 1024 + t] = wv[i] * winv;
}

// ---------------- kernel 6: streaming pass 2 — data partials ----------------
// Grid: B*CHUNKS blocks, 128 threads (one fp32 output column each).
__global__ __launch_bounds__(128) void data_partial_kernel(const float* __restrict__ mem,
                                                            const float* __restrict__ w,
                                                            float* __restrict__ part) {
    const int b = blockIdx.x >> 5;         // CHUNKS = 32
    const int chunk = blockIdx.x & 31;
    const int m = threadIdx.x;
    const float* base = mem + (size_t)b * NN * MM;
    const float* wb = w + (size_t)b * NN;
    const int n0 = chunk * ROWS_PER_CHUNK;
    float acc = 0.f;
    for (int i = 0; i < ROWS_PER_CHUNK; ++i) {
        const int n = n0 + i;
        const float wn = wb[n];             // uniform across block -> scalar broadcast
        acc += wn * __builtin_nontemporal_load(base + (size_t)n * MM + m);
    }
    part[(size_t)blockIdx.x * MM + m] = acc;
}

// ---------------- kernel 7: fixed-order partial reduction -> data ----------------
__global__ __launch_bounds__(256) void data_reduce_kernel(const float* __restrict__ part,
                                                           float* __restrict__ data) {
    const int idx = blockIdx.x * 256 + threadIdx.x;   // B*M = 16384
    const int b = idx >> 7;
    const int m = idx & 127;
    float acc = 0.f;
    for (int c = 0; c < CHUNKS; ++c)
        acc += part[(size_t)(b * CHUNKS + c) * MM + m];
    data[idx] = acc;
}

// ---------------- launch ----------------
extern "C" void kernel_launch(void* const* d_in, const int* in_sizes, int n_in,
                              void* d_out, int out_size, void* d_ws, size_t ws_size,
                              hipStream_t stream) {
    (void)in_sizes; (void)n_in; (void)out_size; (void)ws_size;
    const float* cs  = (const float*)d_in[0];
    const float* pw  = (const float*)d_in[1];
    const float* mem = (const float*)d_in[2];
    const float* Wk  = (const float*)d_in[3];
    const float* bk  = (const float*)d_in[4];
    const float* Wb  = (const float*)d_in[5];
    const float* bb  = (const float*)d_in[6];
    const float* Wg  = (const float*)d_in[7];
    const float* bg  = (const float*)d_in[8];
    const float* Ws  = (const float*)d_in[9];
    const float* bs  = (const float*)d_in[10];
    const float* Wy  = (const float*)d_in[11];
    const float* by  = (const float*)d_in[12];
    // d_in[13] = Wa, d_in[14] = ba : unused by the reference outputs

    float* out_w    = (float*)d_out;                       // (B, N)
    float* out_data = (float*)d_out + (size_t)BB * NN;     // (B, M)

    // Workspace layout (floats)
    float* ws      = (float*)d_ws;
    float* ws_k     = ws;                               // B*M        = 16384
    float* ws_knorm = ws_k     + (size_t)BB * MM;       // B          = 128
    float* ws_beta  = ws_knorm + BB;                    // 128
    float* ws_g     = ws_beta  + BB;                    // 128
    float* ws_y     = ws_g     + BB;                    // 128
    float* ws_s     = ws_y     + BB;                    // 3*B        = 384
    float* ws_cos   = ws_s     + 3 * BB;                // B*N        = 1048576
    float* ws_part  = ws_cos   + (size_t)BB * NN;       // B*CHUNKS*M = 524288

    k_gemm_wmma      <<<64, 32, 0, stream>>>(cs, Wk, bk, ws_k);
    scalars_kernel   <<<BB, 256, 0, stream>>>(cs, Wb, bb, Wg, bg, Ws, bs, Wy, by,
                                              ws_beta, ws_g, ws_s, ws_y);
    knorm_kernel     <<<BB, 32, 0, stream>>>(ws_k, ws_knorm);
    cos_kernel       <<<BB * (NN / 64), 256, 0, stream>>>(mem, ws_k, ws_knorm, ws_cos);
    weights_kernel   <<<BB, 1024, 0, stream>>>(ws_cos, pw, ws_beta, ws_g, ws_s, ws_y, out_w);
    data_partial_kernel<<<BB * CHUNKS, 128, 0, stream>>>(mem, out_w, ws_part);
    data_reduce_kernel <<<(BB * MM) / 256, 256, 0, stream>>>(ws_part, out_data);
}